// VpModel_56710748176864
// MI455X (gfx1250) — compile-verified
//
#include <hip/hip_runtime.h>

// fp32 WMMA fragments (wave32), V_WMMA_F32_16X16X4_F32:
//  A 16x4 f32 : 2 VGPRs/lane. lane holds row m = lane&15; VGPR v holds K = 2*(lane>>4)+v
//  B  4x16 f32: 2 VGPRs/lane. lane holds col n = lane&15; VGPR v holds K = 2*(lane>>4)+v
//  C/D 16x16  : 8 VGPRs/lane. lane holds col n = lane&15; VGPR v holds row m = v + 8*(lane>>4)
typedef __attribute__((ext_vector_type(2))) float v2f;
typedef __attribute__((ext_vector_type(8))) float v8f;

#define SPIRAL_L 9

// Force global address space (1) so loads lower to global_load_* (LOADcnt only),
// never flat_load_* (which also ticks DScnt and forces combined waits).
// Note: use raw ext_vector types here — HIP's float2 class ctor can't bind an
// addrspace(1) reference.
__device__ __forceinline__ float gldf(const float* p) {
  return *(const __attribute__((address_space(1))) float*)p;
}
__device__ __forceinline__ v2f gldf2(const float* p) {
  return *(const __attribute__((address_space(1))) v2f*)p;
}
__device__ __forceinline__ int gldi(const int* p) {
  return *(const __attribute__((address_space(1))) int*)p;
}
__device__ __forceinline__ void gstf(float* p, float v) {
  *(__attribute__((address_space(1))) float*)p = v;
}

__device__ __forceinline__ v8f wmma_f32x4(v2f a, v2f b, v8f c) {
  return __builtin_amdgcn_wmma_f32_16x16x4_f32(false, a, false, b, (short)0, c,
                                               false, false);
}

__device__ __forceinline__ float elu1(float v) {
  return (v > 0.f) ? v : (expf(v) - 1.f);
}

// h[r, co] = elu( sum_k g[r,k] * W[k,co] + bias[co] ),  r = b*Nin + n
// g[r, k]  = x[b, idx[n, k/CIN], k%CIN],  K = L*CIN
// Each wave: 1 row-tile x 2 col-tiles (CO/32 col groups).
template <int CIN, int CO>
__global__ void spiral_elu_t(const float* __restrict__ x,    // (B*Nin) x CIN
                             const int*   __restrict__ idx,  // Nin x L
                             const float* __restrict__ W,    // (L*CIN) x CO
                             const float* __restrict__ bias, // CO
                             float* __restrict__ h,          // (B*Nin) x CO
                             int B, int Nin)
{
  constexpr int K = SPIRAL_L * CIN;
  constexpr int CGROUPS = (CO / 16) / 2;

  const int lane = threadIdx.x & 31;
  const int wave = threadIdx.x >> 5;
  const int R = B * Nin;
  const int rtiles = (R + 15) >> 4;
  const long total = (long)rtiles * CGROUPS;
  const long tile  = (long)blockIdx.x * (blockDim.x >> 5) + wave;
  if (tile >= total) return;                 // wave-uniform: EXEC stays all-ones

  const int rt = (int)(tile / CGROUPS);
  const int cg = (int)(tile % CGROUPS);

  const int mrow  = lane & 15;
  const int khalf = lane >> 4;
  const int col0  = cg * 32 + mrow;          // second col-tile at col0+16

  int r = rt * 16 + mrow;
  if (r >= R) r = R - 1;                     // clamp; dead rows masked at store
  const int bIdx = r / Nin;
  const int n    = r - bIdx * Nin;
  const int* idxrow = idx + n * SPIRAL_L;
  const float* xb   = x + (size_t)bIdx * Nin * CIN;

  v8f acc0 = {0.f, 0.f, 0.f, 0.f, 0.f, 0.f, 0.f, 0.f};
  v8f acc1 = {0.f, 0.f, 0.f, 0.f, 0.f, 0.f, 0.f, 0.f};

  if constexpr ((CIN & 3) == 0) {
    // Fast path: gather pointer computed once per spiral tap l; aligned b64 A loads.
    for (int l = 0; l < SPIRAL_L; ++l) {
      const float* xl = xb + (size_t)gldi(idxrow + l) * CIN + 2 * khalf; // 8B aligned
      const float* wl = W + (size_t)(l * CIN + 2 * khalf) * CO + col0;
#pragma unroll 4
      for (int cb = 0; cb < CIN; cb += 4) {
        v2f a = gldf2(xl);
        v2f b0; b0[0] = gldf(wl);      b0[1] = gldf(wl + CO);
        v2f b1; b1[0] = gldf(wl + 16); b1[1] = gldf(wl + CO + 16);
        acc0 = wmma_f32x4(a, b0, acc0);
        acc1 = wmma_f32x4(a, b1, acc1);
        xl += 4;
        wl += 4 * CO;
      }
    }
  } else {
    // Generic path (CIN=3, K=27): tiny GEMM; /CIN folds to mul-shift, full unroll.
#pragma unroll
    for (int kb = 0; kb < K; kb += 4) {
      v2f a, b0, b1;
#pragma unroll
      for (int v = 0; v < 2; ++v) {
        const int k = kb + 2 * khalf + v;
        float av = 0.f, bv0 = 0.f, bv1 = 0.f;
        if (k < K) {
          const int l = k / CIN;
          const int c = k - l * CIN;
          av  = gldf(xb + (size_t)gldi(idxrow + l) * CIN + c);
          bv0 = gldf(W + (size_t)k * CO + col0);
          bv1 = gldf(W + (size_t)k * CO + col0 + 16);
        }
        a[v] = av; b0[v] = bv0; b1[v] = bv1;
      }
      acc0 = wmma_f32x4(a, b0, acc0);
      acc1 = wmma_f32x4(a, b1, acc1);
    }
  }

  const float bb0 = gldf(bias + col0);
  const float bb1 = gldf(bias + col0 + 16);
#pragma unroll
  for (int v = 0; v < 8; ++v) {
    const int rr = rt * 16 + v + 8 * khalf;
    if (rr < R) {
      float* hp = h + (size_t)rr * CO + col0;
      gstf(hp,      elu1(acc0[v] + bb0));
      gstf(hp + 16, elu1(acc1[v] + bb1));
    }
  }
}

// out[b, m, c] = sum_n D[m, n] * h[b, n, c]
// Each wave: 4 m-tiles x 2 c-tiles (B fragment shared by 4 WMMAs, A by 2).
__global__ void pool_kernel(const float* __restrict__ D,   // Mout x Nin
                            const float* __restrict__ h,   // (B*Nin) x C
                            float* __restrict__ out,       // (B*Mout) x C
                            int B, int Nin, int Mout, int C)
{
  const int lane = threadIdx.x & 31;
  const int wave = threadIdx.x >> 5;
  const int cgroups = C >> 5;                 // 2 c-tiles per group
  const int mgroups = (Mout + 63) >> 6;       // 4 m-tiles per group
  const long perB  = (long)mgroups * cgroups;
  const long total = perB * B;
  const long tile  = (long)blockIdx.x * (blockDim.x >> 5) + wave;
  if (tile >= total) return;                  // wave-uniform

  const int  b  = (int)(tile / perB);
  const long t2 = tile % perB;
  const int  mg = (int)(t2 / cgroups);
  const int  cg = (int)(t2 % cgroups);

  const int mrow  = lane & 15;
  const int khalf = lane >> 4;
  const int col0  = cg * 32 + mrow;

  const float* hb = h + (size_t)b * Nin * C;
  const float* pB = hb + (size_t)(2 * khalf) * C + col0;

  const float* pA[4];
#pragma unroll
  for (int j = 0; j < 4; ++j) {
    int m = mg * 64 + j * 16 + mrow;
    if (m >= Mout) m = Mout - 1;              // clamp; dead rows masked at store
    pA[j] = D + (size_t)m * Nin + 2 * khalf;
  }

  v8f acc[4][2];
#pragma unroll
  for (int j = 0; j < 4; ++j)
#pragma unroll
    for (int t = 0; t < 2; ++t)
      acc[j][t] = (v8f){0.f, 0.f, 0.f, 0.f, 0.f, 0.f, 0.f, 0.f};

  const int Kmain = Nin & ~3;
  for (int kb = 0; kb < Kmain; kb += 4) {
    v2f bf0; bf0[0] = gldf(pB);      bf0[1] = gldf(pB + C);
    v2f bf1; bf1[0] = gldf(pB + 16); bf1[1] = gldf(pB + C + 16);
#pragma unroll
    for (int j = 0; j < 4; ++j) {
      v2f a; a[0] = gldf(pA[j]); a[1] = gldf(pA[j] + 1);
      acc[j][0] = wmma_f32x4(a, bf0, acc[j][0]);
      acc[j][1] = wmma_f32x4(a, bf1, acc[j][1]);
      pA[j] += 4;
    }
    pB += (size_t)4 * C;
  }

  if (Kmain < Nin) {                          // peeled masked tail (once per tile)
    v2f a[4], bf0, bf1;
#pragma unroll
    for (int v = 0; v < 2; ++v) {
      const int nk = Kmain + 2 * khalf + v;
      const bool ok = (nk < Nin);
      bf0[v] = ok ? gldf(hb + (size_t)nk * C + col0)      : 0.f;
      bf1[v] = ok ? gldf(hb + (size_t)nk * C + col0 + 16) : 0.f;
#pragma unroll
      for (int j = 0; j < 4; ++j)
        a[j][v] = ok ? gldf(pA[j] + v) : 0.f;
    }
#pragma unroll
    for (int j = 0; j < 4; ++j) {
      acc[j][0] = wmma_f32x4(a[j], bf0, acc[j][0]);
      acc[j][1] = wmma_f32x4(a[j], bf1, acc[j][1]);
    }
  }

#pragma unroll
  for (int j = 0; j < 4; ++j) {
    const int mbase = mg * 64 + j * 16;
#pragma unroll
    for (int v = 0; v < 8; ++v) {
      const int mm = mbase + v + 8 * khalf;
      if (mm < Mout) {
        float* op = out + ((size_t)b * Mout + mm) * C + col0;
        gstf(op,      acc[j][0][v]);
        gstf(op + 16, acc[j][1][v]);
      }
    }
  }
}

// out[r, c] = sum_k g[r, k] * Wfc[k, c] + bfc[c]   (256 x 5120 x 128, all aligned)
__global__ void fc_kernel(const float* __restrict__ g,
                          const float* __restrict__ Wfc,
                          const float* __restrict__ bfc,
                          float* __restrict__ out,
                          int Brows, int K, int Ncols)
{
  const int lane = threadIdx.x & 31;
  const int wave = threadIdx.x >> 5;
  const int ctiles = Ncols >> 4;
  const int rtiles = Brows >> 4;
  const long total = (long)rtiles * ctiles;
  const long tile  = (long)blockIdx.x * (blockDim.x >> 5) + wave;
  if (tile >= total) return;                  // wave-uniform

  const int rt = (int)(tile / ctiles);
  const int ct = (int)(tile % ctiles);

  const int mrow  = lane & 15;
  const int khalf = lane >> 4;
  const int col   = ct * 16 + mrow;

  const float* pA = g + (size_t)(rt * 16 + mrow) * K + 2 * khalf;  // 8B aligned
  const float* pW = Wfc + (size_t)(2 * khalf) * Ncols + col;

  v8f acc = {0.f, 0.f, 0.f, 0.f, 0.f, 0.f, 0.f, 0.f};
  for (int kb = 0; kb < K; kb += 4) {
    v2f a = gldf2(pA);
    v2f b; b[0] = gldf(pW); b[1] = gldf(pW + Ncols);
    acc = wmma_f32x4(a, b, acc);
    pA += 4;
    pW += (size_t)4 * Ncols;
  }

  const float bb = gldf(bfc + col);
#pragma unroll
  for (int v = 0; v < 8; ++v) {
    const int rr = rt * 16 + v + 8 * khalf;
    gstf(out + (size_t)rr * Ncols + col, acc[v] + bb);
  }
}

extern "C" void kernel_launch(void* const* d_in, const int* in_sizes, int n_in,
                              void* d_out, int out_size, void* d_ws, size_t ws_size,
                              hipStream_t stream) {
  (void)in_sizes; (void)n_in; (void)out_size; (void)ws_size;

  // setup_inputs() dict order: x, (idx,D,W,b) x4, Wfc, bfc
  const int*   idxs[4] = {(const int*)d_in[1], (const int*)d_in[5],
                          (const int*)d_in[9], (const int*)d_in[13]};
  const float* Ds[4]   = {(const float*)d_in[2], (const float*)d_in[6],
                          (const float*)d_in[10], (const float*)d_in[14]};
  const float* Ws[4]   = {(const float*)d_in[3], (const float*)d_in[7],
                          (const float*)d_in[11], (const float*)d_in[15]};
  const float* bs[4]   = {(const float*)d_in[4], (const float*)d_in[8],
                          (const float*)d_in[12], (const float*)d_in[16]};
  const float* x   = (const float*)d_in[0];
  const float* Wfc = (const float*)d_in[17];
  const float* bfc = (const float*)d_in[18];

  const int B = 256;
  const int Ns[5] = {5023, 1256, 314, 79, 20};
  const int Cs[5] = {3, 32, 64, 128, 256};

  // Workspace ping-pong: bufH = spiral outputs (max B*5023*32 floats ~165MB),
  // bufG = pooled outputs (max B*1256*32 floats ~41MB).
  float* bufH = (float*)d_ws;
  float* bufG = bufH + (size_t)B * 5023 * 32;

  const int WAVES = 4;                        // 128 threads / block
  const float* cur = x;
  for (int i = 0; i < 4; ++i) {
    const int Nin = Ns[i], Mout = Ns[i + 1], Co = Cs[i + 1];

    const long R = (long)B * Nin;
    const long t1 = ((R + 15) / 16) * (Co / 32);
    const unsigned g1 = (unsigned)((t1 + WAVES - 1) / WAVES);
    switch (i) {
      case 0: spiral_elu_t<3, 32><<<dim3(g1), dim3(32 * WAVES), 0, stream>>>(
                  cur, idxs[0], Ws[0], bs[0], bufH, B, Nin); break;
      case 1: spiral_elu_t<32, 64><<<dim3(g1), dim3(32 * WAVES), 0, stream>>>(
                  cur, idxs[1], Ws[1], bs[1], bufH, B, Nin); break;
      case 2: spiral_elu_t<64, 128><<<dim3(g1), dim3(32 * WAVES), 0, stream>>>(
                  cur, idxs[2], Ws[2], bs[2], bufH, B, Nin); break;
      case 3: spiral_elu_t<128, 256><<<dim3(g1), dim3(32 * WAVES), 0, stream>>>(
                  cur, idxs[3], Ws[3], bs[3], bufH, B, Nin); break;
    }

    const long t2 = (long)B * ((Mout + 63) / 64) * (Co / 32);
    const unsigned g2 = (unsigned)((t2 + WAVES - 1) / WAVES);
    pool_kernel<<<dim3(g2), dim3(32 * WAVES), 0, stream>>>(
        Ds[i], bufH, bufG, B, Nin, Mout, Co);

    cur = bufG;
  }

  // Final FC: (256 x 5120) @ (5120 x 128) + bias
  const int Kfc = Ns[4] * Cs[4];              // 5120
  const long t3 = (256 / 16) * (128 / 16);
  const unsigned g3 = (unsigned)((t3 + WAVES - 1) / WAVES);
  fc_kernel<<<dim3(g3), dim3(32 * WAVES), 0, stream>>>(
      bufG, Wfc, bfc, (float*)d_out, 256, Kfc, 128);
}